// RankBasedAveragePoolingLayer_45337674777262
// MI455X (gfx1250) — compile-verified
//
#include <hip/hip_runtime.h>

// Rank-based 2x2 average pooling (mean of top-k of each 2x2 window, per channel)
// Input : x (B=16, H=128, W=128, C=256) float32, NHWC
// Output: (B, H/2, W/2, C) float32
//
// Pure HBM-streaming kernel: 320 MiB total traffic -> ~14 us floor at 23.3 TB/s.
// No matmul structure => WMMA inapplicable; direct coalesced b128 loads with
// non-temporal hints (single-touch stream larger than the 192 MB L2) are the
// optimal data path on gfx1250.

typedef float v4f __attribute__((ext_vector_type(4)));

constexpr int B  = 16;
constexpr int H  = 128;
constexpr int W  = 128;
constexpr int C  = 256;
constexpr int HO = H / 2;            // 64
constexpr int WO = W / 2;            // 64
constexpr int C4 = C / 4;            // 64 float4 groups per pixel
constexpr long TOTAL_V4 = (long)B * HO * WO * C4;   // 4,194,304 threads

__device__ __forceinline__ void cswap_desc(float& p, float& q) {
    const float hi = fmaxf(p, q);
    const float lo = fminf(p, q);
    p = hi;
    q = lo;
}

__global__ __launch_bounds__(256) void rank_pool2x2_kernel(
        const float* __restrict__ x,
        const int*   __restrict__ kptr,
        float*       __restrict__ out) {
    const int tid = blockIdx.x * blockDim.x + threadIdx.x;

    // decode (b, ho, wo, c4) -- all power-of-two dims, 6 bits each for c4/wo/ho
    const int c4 = tid & (C4 - 1);
    const int wo = (tid >> 6)  & (WO - 1);
    const int ho = (tid >> 12) & (HO - 1);
    const int b  = tid >> 18;

    // uniform scalar threshold (s_load); clamp to [1,4] for safety
    int k = kptr[0];
    k = k < 1 ? 1 : (k > 4 ? 4 : k);
    const float rk = 1.0f / (float)k;

    // input float offset: ((b*H + 2*ho)*W + 2*wo)*C + 4*c4
    const size_t in_base =
        (((size_t)(b * H + 2 * ho) * W) + (size_t)(2 * wo)) * C + (size_t)c4 * 4;
    const v4f* p = (const v4f*)(x + in_base);

    // four window taps: +0, +C floats (next column), +W*C floats (next row)
    const v4f r0c0 = __builtin_nontemporal_load(p);
    const v4f r0c1 = __builtin_nontemporal_load(p + C4);
    const v4f r1c0 = __builtin_nontemporal_load(p + (size_t)W * C4);
    const v4f r1c1 = __builtin_nontemporal_load(p + (size_t)W * C4 + C4);

    v4f res;
#pragma unroll
    for (int i = 0; i < 4; ++i) {
        float s0 = r0c0[i];
        float s1 = r0c1[i];
        float s2 = r1c0[i];
        float s3 = r1c1[i];
        // 5-comparator descending sort network for 4 elements
        cswap_desc(s0, s1);
        cswap_desc(s2, s3);
        cswap_desc(s0, s2);
        cswap_desc(s1, s3);
        cswap_desc(s1, s2);
        // mean of top-k (uniform k -> scalar-branch / cndmask)
        float sum = s0;
        if (k > 1) sum += s1;
        if (k > 2) sum += s2;
        if (k > 3) sum += s3;
        res[i] = sum * rk;
    }

    __builtin_nontemporal_store(res, (v4f*)out + tid);
}

extern "C" void kernel_launch(void* const* d_in, const int* in_sizes, int n_in,
                              void* d_out, int out_size, void* d_ws, size_t ws_size,
                              hipStream_t stream) {
    const float* x    = (const float*)d_in[0];
    const int*   kthr = (const int*)d_in[1];
    float*       out  = (float*)d_out;

    const int  block = 256;                       // 8 wave32 per block
    const long grid  = TOTAL_V4 / block;          // 16384 blocks

    rank_pool2x2_kernel<<<(unsigned)grid, block, 0, stream>>>(x, kthr, out);

    (void)in_sizes; (void)n_in; (void)out_size; (void)d_ws; (void)ws_size;
}